// CorrMLP_82901458747941
// MI455X (gfx1250) — compile-verified
//
#include <hip/hip_runtime.h>
#include <hip/hip_bf16.h>
#include <math.h>

typedef __attribute__((ext_vector_type(2))) float v2f;
typedef __attribute__((ext_vector_type(8))) float v8f;

#define SDIM 64
#define SP (SDIM*SDIM*SDIM)   // 262144 spatial points
#define CC 27
#define LN_EPS 1e-5f

__device__ __forceinline__ float gelu_exact(float x) {
    return 0.5f * x * (1.0f + erff(x * 0.70710678118654752440f));
}

// ---------------------------------------------------------------------------
// Kernel 1: max_disp=1 cost volume, channel-mean, channel-last output.
// Block = 4x4x16 tile of points; x2 halo (6x6x18) staged in LDS per channel,
// double-buffered, so the 27 taps are LDS reads instead of global loads.
// corr[s*27 + o], o = (dx+1)*9 + (dy+1)*3 + (dz+1)
// ---------------------------------------------------------------------------
#define HLX 6
#define HLY 6
#define HLZ 18
#define HALO_N (HLX*HLY*HLZ)   // 648

__global__ void __launch_bounds__(256) corr_kernel(const float* __restrict__ x1,
                                                   const float* __restrict__ x2,
                                                   float* __restrict__ corr) {
    __shared__ float halo[2][HALO_N];
    const int tid = threadIdx.x;
    // local point coords within tile (z fastest for coalescing)
    const int lz = tid & 15;
    const int ly = (tid >> 4) & 3;
    const int lx = tid >> 6;
    // tile origin
    const int bz = (blockIdx.x & 3) << 4;         // 4 z-tiles of 16
    const int by = ((blockIdx.x >> 2) & 15) << 2; // 16 y-tiles of 4
    const int bx = (blockIdx.x >> 6) << 2;        // 16 x-tiles of 4
    const int x = bx + lx, y = by + ly, z = bz + lz;
    const int s = (x << 12) + (y << 6) + z;

    float acc[27];
#pragma unroll
    for (int o = 0; o < 27; ++o) acc[o] = 0.f;

    // preload channel 0 halo
    {
        const float* b2 = x2;
        for (int i = tid; i < HALO_N; i += 256) {
            int hz = i % HLZ, r = i / HLZ, hy = r % HLY, hx = r / HLY;
            int gx = bx + hx - 1, gy = by + hy - 1, gz = bz + hz - 1;
            bool ok = ((unsigned)gx < 64u) && ((unsigned)gy < 64u) && ((unsigned)gz < 64u);
            halo[0][i] = ok ? b2[(gx << 12) + (gy << 6) + gz] : 0.f;
        }
    }
    for (int ch = 0; ch < 64; ++ch) {
        __syncthreads();
        // prefetch next channel's halo into the other buffer
        if (ch + 1 < 64) {
            const float* b2 = x2 + (size_t)(ch + 1) * SP;
            float* dst = halo[(ch + 1) & 1];
            for (int i = tid; i < HALO_N; i += 256) {
                int hz = i % HLZ, r = i / HLZ, hy = r % HLY, hx = r / HLY;
                int gx = bx + hx - 1, gy = by + hy - 1, gz = bz + hz - 1;
                bool ok = ((unsigned)gx < 64u) && ((unsigned)gy < 64u) && ((unsigned)gz < 64u);
                dst[i] = ok ? b2[(gx << 12) + (gy << 6) + gz] : 0.f;
            }
        }
        const float v1 = x1[(size_t)ch * SP + s];
        const float* hb = halo[ch & 1];
        int o = 0;
#pragma unroll
        for (int dx = 0; dx < 3; ++dx)
#pragma unroll
            for (int dy = 0; dy < 3; ++dy)
#pragma unroll
                for (int dz = 0; dz < 3; ++dz, ++o)
                    acc[o] = fmaf(v1, hb[((lx + dx) * HLY + (ly + dy)) * HLZ + (lz + dz)], acc[o]);
    }
    float* out = corr + (size_t)s * 27;
#pragma unroll
    for (int o = 0; o < 27; ++o) out[o] = acc[o] * (1.0f / 64.0f);
}

// ---------------------------------------------------------------------------
// Kernel 2: per-point LayerNorm over the 27 channels
// ---------------------------------------------------------------------------
__global__ void __launch_bounds__(256) preln_kernel(const float* __restrict__ xin,
                                                    const float* __restrict__ g,
                                                    const float* __restrict__ b,
                                                    float* __restrict__ xln) {
    int s = blockIdx.x * blockDim.x + threadIdx.x;
    if (s >= SP) return;
    const float* r = xin + (size_t)s * 27;
    float vals[27];
    float m = 0.f;
#pragma unroll
    for (int c = 0; c < 27; ++c) { vals[c] = r[c]; m += vals[c]; }
    m *= (1.f / 27.f);
    float var = 0.f;
#pragma unroll
    for (int c = 0; c < 27; ++c) { float d = vals[c] - m; var += d * d; }
    var *= (1.f / 27.f);
    float rs = rsqrtf(var + LN_EPS);
    float* o = xln + (size_t)s * 27;
#pragma unroll
    for (int c = 0; c < 27; ++c) o[c] = (vals[c] - m) * rs * g[c] + b[c];
}

// ---------------------------------------------------------------------------
// Kernel 2b: transpose + zero-pad sgu weights for unguarded, vectorized
// A-fragment loads:  Wt[q][k] = sgu_W[k][q]  (Pp x Pp, zeros beyond P)
//                    bpad[q]  = sgu_b[q]     (zeros beyond P)
// ---------------------------------------------------------------------------
__global__ void __launch_bounds__(256) padw_kernel(const float* __restrict__ W,
                                                   const float* __restrict__ bias,
                                                   float* __restrict__ Wt,
                                                   float* __restrict__ bpad,
                                                   int P, int Pp) {
    int i = blockIdx.x * blockDim.x + threadIdx.x;
    int tot = Pp * Pp;
    if (i < tot) {
        int q = i / Pp, k = i - q * Pp;
        Wt[i] = (q < P && k < P) ? W[k * P + q] : 0.f;
    }
    if (i < Pp) bpad[i] = (i < P) ? bias[i] : 0.f;
}

// ---------------------------------------------------------------------------
// Kernel 3: windowed gated MLP; one workgroup (8 waves) per f^3 window.
// Dynamic LDS layout (floats):
//   t  [Pp][32]  LN'ed window input -> later reused for gated activation
//   u  [Pp][28]  gelu "u" half
//   v  [Pp][32]  gelu "v" half, then row-LN'ed (zero-padded cols/rows)
//   wb [2048]    staged weight tile (in_W 32x64, then out_W 32x32)
// All three linears run on V_WMMA_F32_16X16X4_F32.
// ---------------------------------------------------------------------------
__global__ void __launch_bounds__(256) wingmlp_kernel(
    const float* __restrict__ xln, float* __restrict__ xout,
    const float* __restrict__ wln_g, const float* __restrict__ wln_b,
    const float* __restrict__ inW,  const float* __restrict__ inB,    // [27][54],[54]
    const float* __restrict__ sln_g,const float* __restrict__ sln_b,
    const float* __restrict__ Wt,   const float* __restrict__ sguBp,  // [Pp][Pp],[Pp] padded
    const float* __restrict__ outW, const float* __restrict__ outB,   // [27][27],[27]
    int f, int P, int Pp, int gdim)
{
    extern __shared__ float lds[];
    float* t  = lds;                 // Pp*32
    float* u  = lds + Pp * 32;       // Pp*28
    float* v  = u + Pp * 28;         // Pp*32
    float* wb = v + Pp * 32;         // 2048

    const int tid  = threadIdx.x;
    const int lane = tid & 31;
    const int wave = tid >> 5;
    const int hi   = lane >> 4;      // half-wave select (K/M split per ISA layout)
    const int l15  = lane & 15;
    const int ff   = f * f;
    const int Mt   = Pp >> 4;

    int g  = blockIdx.x;
    int gz = g % gdim;
    int gy = (g / gdim) % gdim;
    int gx = g / (gdim * gdim);
    int bx = gx * f, by = gy * f, bz = gz * f;

    // ---- stage 0: zero t/u/v, stage in_W into 32x64 zero-padded tile ----
    for (int i = tid; i < Pp * 92; i += 256) lds[i] = 0.f;
    for (int i = tid; i < 2048; i += 256) {
        int k = i >> 6, n = i & 63;
        wb[i] = (k < 27 && n < 54) ? inW[k * 54 + n] : 0.f;
    }
    __syncthreads();

    // ---- stage 1: gather window (zero for out-of-range) + window LayerNorm ----
    for (int p = tid; p < P; p += 256) {
        int ih = p / ff, rem = p % ff, iw = rem / f, id = rem % f;
        int sx = bx + ih, sy = by + iw, sz = bz + id;
        bool valid = (sx < SDIM) && (sy < SDIM) && (sz < SDIM);
        const float* row = xln + (size_t)((sx << 12) + (sy << 6) + sz) * 27;
        float vals[27];
        float m = 0.f;
#pragma unroll
        for (int c = 0; c < 27; ++c) { vals[c] = valid ? row[c] : 0.f; m += vals[c]; }
        m *= (1.f / 27.f);
        float var = 0.f;
#pragma unroll
        for (int c = 0; c < 27; ++c) { float d = vals[c] - m; var += d * d; }
        var *= (1.f / 27.f);
        float rs = rsqrtf(var + LN_EPS);
#pragma unroll
        for (int c = 0; c < 27; ++c)
            t[p * 32 + c] = (vals[c] - m) * rs * wln_g[c] + wln_b[c];
    }
    __syncthreads();

    // ---- stage 2: in-projection  D[Pp x 64] = t[Pp x 32] @ wb[32 x 64], gelu, split u|v
    for (int tile = wave; tile < Mt * 4; tile += 8) {
        int mt = tile >> 2, nt = tile & 3;
        v8f acc = {};
        int row = mt * 16 + l15;
#pragma unroll
        for (int kk = 0; kk < 8; ++kk) {
            int k = kk * 4 + hi * 2;
            v2f a, b;
            a[0] = t[row * 32 + k];
            a[1] = t[row * 32 + k + 1];
            b[0] = wb[k * 64 + nt * 16 + l15];
            b[1] = wb[(k + 1) * 64 + nt * 16 + l15];
            acc = __builtin_amdgcn_wmma_f32_16x16x4_f32(false, a, false, b,
                                                        (short)0, acc, false, false);
        }
        int n = nt * 16 + l15;
        if (n < 54) {
            float bias = inB[n];
#pragma unroll
            for (int i = 0; i < 8; ++i) {
                int q = mt * 16 + i + hi * 8;
                if (q < P) {
                    float h = gelu_exact(acc[i] + bias);
                    if (n < 27) u[q * 28 + n] = h;
                    else        v[q * 32 + (n - 27)] = h;
                }
            }
        }
    }
    __syncthreads();

    // ---- stage 3: LayerNorm each valid v row; restage wb with out_W (32x32) ----
    for (int p = tid; p < P; p += 256) {
        float vals[27];
        float m = 0.f;
#pragma unroll
        for (int c = 0; c < 27; ++c) { vals[c] = v[p * 32 + c]; m += vals[c]; }
        m *= (1.f / 27.f);
        float var = 0.f;
#pragma unroll
        for (int c = 0; c < 27; ++c) { float d = vals[c] - m; var += d * d; }
        var *= (1.f / 27.f);
        float rs = rsqrtf(var + LN_EPS);
#pragma unroll
        for (int c = 0; c < 27; ++c)
            v[p * 32 + c] = (vals[c] - m) * rs * sln_g[c] + sln_b[c];
    }
    for (int i = tid; i < 1024; i += 256) {
        int k = i >> 5, n = i & 31;
        wb[i] = (k < 27 && n < 27) ? outW[k * 27 + n] : 0.f;
    }
    __syncthreads();

    // ---- stage 4: spatial gate  vq[q,c] = sum_p Wt[q][p] * v[p][c]; gate into t
    // Wt is pre-transposed + zero padded -> unguarded v2f (b64) global loads.
    int Kt = Pp >> 2;
    for (int tile = wave; tile < Mt * 2; tile += 8) {
        int mt = tile >> 1, nt = tile & 1;
        v8f acc = {};
        int q0 = mt * 16 + l15;
        const v2f* wrow = (const v2f*)(Wt + (size_t)q0 * Pp);
        const float* vcol = v + nt * 16 + l15;
        for (int kk = 0; kk < Kt; ++kk) {
            int k = kk * 4 + hi * 2;
            v2f a = wrow[k >> 1];
            v2f b;
            b[0] = vcol[k * 32];
            b[1] = vcol[(k + 1) * 32];
            acc = __builtin_amdgcn_wmma_f32_16x16x4_f32(false, a, false, b,
                                                        (short)0, acc, false, false);
        }
        int c = nt * 16 + l15;
        if (c < 27) {
#pragma unroll
            for (int i = 0; i < 8; ++i) {
                int q = mt * 16 + i + hi * 8;
                if (q < P)
                    t[q * 32 + c] = (acc[i] + sguBp[q] + 1.0f) * u[q * 28 + c];
            }
        }
    }
    __syncthreads();

    // ---- stage 5: out-projection + residual, scatter valid positions ----
    for (int tile = wave; tile < Mt * 2; tile += 8) {
        int mt = tile >> 1, nt = tile & 1;
        v8f acc = {};
        int row = mt * 16 + l15;
#pragma unroll
        for (int kk = 0; kk < 8; ++kk) {
            int k = kk * 4 + hi * 2;
            v2f a, b;
            a[0] = t[row * 32 + k];
            a[1] = t[row * 32 + k + 1];
            b[0] = wb[k * 32 + nt * 16 + l15];
            b[1] = wb[(k + 1) * 32 + nt * 16 + l15];
            acc = __builtin_amdgcn_wmma_f32_16x16x4_f32(false, a, false, b,
                                                        (short)0, acc, false, false);
        }
        int c = nt * 16 + l15;
        if (c < 27) {
            float bias = outB[c];
#pragma unroll
            for (int i = 0; i < 8; ++i) {
                int q = mt * 16 + i + hi * 8;
                if (q < P) {
                    int ih = q / ff, rem = q % ff, iw = rem / f, id = rem % f;
                    int sx = bx + ih, sy = by + iw, sz = bz + id;
                    if (sx < SDIM && sy < SDIM && sz < SDIM) {
                        size_t sp = (size_t)((sx << 12) + (sy << 6) + sz);
                        xout[sp * 27 + c] = acc[i] + bias + xln[sp * 27 + c];
                    }
                }
            }
        }
    }
}

// ---------------------------------------------------------------------------
// Kernel 4a/4b: spatial mean of (x1+x2+x3) -> asum[27] (block-reduced atomics)
// ---------------------------------------------------------------------------
__global__ void zero_kernel(float* __restrict__ p, int n) {
    int i = blockIdx.x * blockDim.x + threadIdx.x;
    if (i < n) p[i] = 0.f;
}

__global__ void __launch_bounds__(256) sum3_kernel(const float* __restrict__ a,
                                                   const float* __restrict__ b,
                                                   const float* __restrict__ c,
                                                   float* __restrict__ asum) {
    __shared__ float bs[27];
    int tid = threadIdx.x;
    if (tid < 27) bs[tid] = 0.f;
    __syncthreads();
    float loc[27];
#pragma unroll
    for (int k = 0; k < 27; ++k) loc[k] = 0.f;
    for (int s = blockIdx.x * blockDim.x + tid; s < SP; s += gridDim.x * blockDim.x) {
        size_t o = (size_t)s * 27;
#pragma unroll
        for (int k = 0; k < 27; ++k) loc[k] += a[o + k] + b[o + k] + c[o + k];
    }
#pragma unroll
    for (int k = 0; k < 27; ++k) atomicAdd(&bs[k], loc[k]);
    __syncthreads();
    if (tid < 27) atomicAdd(&asum[tid], bs[tid]);
}

// ---------------------------------------------------------------------------
// Kernel 5: reweight MLP (27 -> 6 gelu -> 81) + softmax over the 3 windows
// wgt[k*27 + ci] = softmax_k( z[ci*3 + k] )
// ---------------------------------------------------------------------------
__global__ void reweight_kernel(const float* __restrict__ asum,
                                const float* __restrict__ rw1W, const float* __restrict__ rw1b,
                                const float* __restrict__ rw2W, const float* __restrict__ rw2b,
                                float* __restrict__ wgt) {
    __shared__ float a[27], h[6], zz[81];
    int tid = threadIdx.x;
    if (tid < 27) a[tid] = asum[tid] * (1.0f / (float)SP);
    __syncthreads();
    if (tid < 6) {
        float s = rw1b[tid];
        for (int c = 0; c < 27; ++c) s += a[c] * rw1W[c * 6 + tid];
        h[tid] = gelu_exact(s);
    }
    __syncthreads();
    if (tid < 81) {
        float s = rw2b[tid];
        for (int j = 0; j < 6; ++j) s += h[j] * rw2W[j * 81 + tid];
        zz[tid] = s;
    }
    __syncthreads();
    if (tid < 27) {
        float z0 = zz[tid * 3], z1 = zz[tid * 3 + 1], z2 = zz[tid * 3 + 2];
        float mx = fmaxf(z0, fmaxf(z1, z2));
        float e0 = expf(z0 - mx), e1 = expf(z1 - mx), e2 = expf(z2 - mx);
        float inv = 1.f / (e0 + e1 + e2);
        wgt[tid]      = e0 * inv;
        wgt[27 + tid] = e1 * inv;
        wgt[54 + tid] = e2 * inv;
    }
}

// ---------------------------------------------------------------------------
// Kernel 6: weighted blend + final 27x27 linear + correlation residual
// ---------------------------------------------------------------------------
__global__ void __launch_bounds__(256) final_kernel(const float* __restrict__ xa,
                                                    const float* __restrict__ xb,
                                                    const float* __restrict__ xc,
                                                    const float* __restrict__ wgt,
                                                    const float* __restrict__ corr,
                                                    const float* __restrict__ oW,
                                                    const float* __restrict__ oB,
                                                    float* __restrict__ out) {
    int s = blockIdx.x * blockDim.x + threadIdx.x;
    if (s >= SP) return;
    size_t o = (size_t)s * 27;
    float xx[27];
#pragma unroll
    for (int c = 0; c < 27; ++c)
        xx[c] = xa[o + c] * wgt[c] + xb[o + c] * wgt[27 + c] + xc[o + c] * wgt[54 + c];
#pragma unroll
    for (int c = 0; c < 27; ++c) {
        float acc = oB[c] + corr[o + c];
#pragma unroll
        for (int k = 0; k < 27; ++k) acc = fmaf(xx[k], oW[k * 27 + c], acc);
        out[o + c] = acc;
    }
}

// ---------------------------------------------------------------------------
// Host launcher
// ---------------------------------------------------------------------------
extern "C" void kernel_launch(void* const* d_in, const int* in_sizes, int n_in,
                              void* d_out, int out_size, void* d_ws, size_t ws_size,
                              hipStream_t stream) {
    auto F = [&](int i) { return (const float*)d_in[i]; };
    const float* x1   = F(0);
    const float* x2   = F(1);
    const float* ln_g = F(2);
    const float* ln_b = F(3);
    // window param blocks: ln_g,ln_b,in_W,in_b,sgu_ln_g,sgu_ln_b,sgu_W,sgu_b,out_W,out_b
    const int W3 = 4, W5 = 14, W7 = 24;
    const float* rw1W = F(34);
    const float* rw1b = F(35);
    const float* rw2W = F(36);
    const float* rw2b = F(37);
    const float* foW  = F(38);
    const float* foB  = F(39);

    float* ws = (float*)d_ws;
    const size_t N27 = (size_t)SP * 27;
    float* corr = ws;
    float* xln  = ws + N27;
    float* xf1  = ws + 2 * N27;
    float* xf2  = ws + 3 * N27;
    float* xf3  = ws + 4 * N27;
    float* asum = ws + 5 * N27;          // 32 floats
    float* wgt  = asum + 32;             // 96 floats (81 used)
    float* b3   = wgt + 96;              // 32
    float* b5   = b3 + 32;               // 128
    float* b7   = b5 + 128;              // 352
    float* Wt3  = b7 + 352;              // 32*32
    float* Wt5  = Wt3 + 32 * 32;         // 128*128
    float* Wt7  = Wt5 + 128 * 128;       // 352*352

    corr_kernel<<<SP / 256, 256, 0, stream>>>(x1, x2, corr);
    preln_kernel<<<SP / 256, 256, 0, stream>>>(corr, ln_g, ln_b, xln);

    // pad/transpose sgu weights (tiny, L2-resident, reused by every window)
    padw_kernel<<<(32 * 32 + 255) / 256, 256, 0, stream>>>(F(W3 + 6), F(W3 + 7), Wt3, b3, 27, 32);
    padw_kernel<<<(128 * 128 + 255) / 256, 256, 0, stream>>>(F(W5 + 6), F(W5 + 7), Wt5, b5, 125, 128);
    padw_kernel<<<(352 * 352 + 255) / 256, 256, 0, stream>>>(F(W7 + 6), F(W7 + 7), Wt7, b7, 343, 352);

    // f=3: P=27, Pp=32, gdim=22 ; f=5: P=125, Pp=128, gdim=13 ; f=7: P=343, Pp=352, gdim=10
    auto launch_win = [&](int base, float* xout, const float* Wt, const float* bp,
                          int f, int P, int Pp, int gdim) {
        size_t shmem = (size_t)(Pp * 92 + 2048) * sizeof(float);
        wingmlp_kernel<<<gdim * gdim * gdim, 256, shmem, stream>>>(
            xln, xout,
            F(base + 0), F(base + 1), F(base + 2), F(base + 3),
            F(base + 4), F(base + 5), Wt, bp,
            F(base + 8), F(base + 9),
            f, P, Pp, gdim);
    };
    launch_win(W3, xf1, Wt3, b3, 3, 27, 32, 22);
    launch_win(W5, xf2, Wt5, b5, 5, 125, 128, 13);
    launch_win(W7, xf3, Wt7, b7, 7, 343, 352, 10);

    zero_kernel<<<1, 32, 0, stream>>>(asum, 27);
    sum3_kernel<<<256, 256, 0, stream>>>(xf1, xf2, xf3, asum);
    reweight_kernel<<<1, 128, 0, stream>>>(asum, rw1W, rw1b, rw2W, rw2b, wgt);
    final_kernel<<<SP / 256, 256, 0, stream>>>(xf1, xf2, xf3, wgt, corr, foW, foB,
                                               (float*)d_out);
}